// MPS_31361851195851
// MI455X (gfx1250) — compile-verified
//
#include <hip/hip_runtime.h>

typedef float v2f __attribute__((ext_vector_type(2)));
typedef float v8f __attribute__((ext_vector_type(8)));

#define B_ 128
#define N_ 1024
#define D_ 20
#define F_ 2
#define O_ 10
#define L_ 512
#define SITES_PER_BLK 32
#define NBLK 16           // blocks per side
#define WAVES 16
#define THREADS (WAVES * 32)

// Compiler-only fence: LDS ops within a wave are executed in order by the DS
// pipeline (CDNA5 ISA 7.1), and the compiler inserts s_wait_dscnt before uses;
// we only need to stop the compiler from reordering around the round trip.
#define WAVE_LDS_FENCE() asm volatile("" ::: "memory")

// Entry (k,n) of the padded 32x32 site matrix M = I + A0*x0 + A1*x1 (zero outside 20x20)
__device__ __forceinline__ float site_entry(const float* __restrict__ Ts,
                                            int k, int n, float x0, float x1) {
  float v = 0.f;
  if (k < D_ && n < D_) {
    const float2 e = *(const float2*)(Ts + (k * D_ + n) * F_);
    v = fmaf(e.x, x0, e.y * x1);
    if (k == n) v += 1.f;
  }
  return v;
}

__global__ __launch_bounds__(THREADS)
void mps_fused(const float* __restrict__ inp,   // [B,N,F]
               const float* __restrict__ lt,    // [L,D,D,F]
               const float* __restrict__ ot,    // [O,D,D]
               const float* __restrict__ rt,    // [N-L,D,D,F]
               float* __restrict__ out) {       // [B,O]
  __shared__ __align__(16) float scr[WAVES][32 * 32];       // 64 KB: per-wave A-layout round trip
  __shared__ __align__(16) float prod[2 * NBLK][D_ * 32];   // 80 KB: block products (rows<20)

  const int tid   = threadIdx.x;
  const int w     = tid >> 5;
  const int lane  = tid & 31;
  const int lhalf = lane >> 4;   // 0 or 1
  const int l16   = lane & 15;
  const int b     = blockIdx.x;

  float* S = scr[w];

  for (int side = 0; side < 2; ++side) {
    const float* T = side ? rt : lt;
    const int sbase = w * SITES_PER_BLK;
    v8f c[2][2];   // current 32x32 chain product P, C/D layout tiles

    for (int sit = 0; sit < SITES_PER_BLK; ++sit) {
      const int sl = sbase + sit;
      const float2 x =
          *(const float2*)(inp + ((size_t)b * N_ + (size_t)side * L_ + sl) * F_);
      const float* Ts = T + (size_t)sl * (D_ * D_ * F_);

      // Build B fragments of M (B layout: VGPR j holds row kb+j, col = tj*16+l16)
      v2f bf[2][5];
#pragma unroll
      for (int tj = 0; tj < 2; ++tj) {
#pragma unroll
        for (int kk = 0; kk < 5; ++kk) {
          const int n  = tj * 16 + l16;
          const int kb = kk * 4 + 2 * lhalf;
          v2f f;
          f.x = site_entry(Ts, kb + 0, n, x.x, x.y);
          f.y = site_entry(Ts, kb + 1, n, x.x, x.y);
          bf[tj][kk] = f;
        }
      }

      if (sit == 0) {
        // P = M directly in C/D layout
#pragma unroll
        for (int ti = 0; ti < 2; ++ti)
#pragma unroll
          for (int tj = 0; tj < 2; ++tj)
#pragma unroll
            for (int v = 0; v < 8; ++v) {
              const int m = ti * 16 + v + 8 * lhalf;
              const int n = tj * 16 + l16;
              c[ti][tj][v] = site_entry(Ts, m, n, x.x, x.y);
            }
      } else {
        // Spill P row-major to wave-private LDS so it can be reloaded in
        // A-operand layout. No workgroup barrier needed: same-wave DS ops are
        // in-order in hardware; fence only stops compiler reordering.
#pragma unroll
        for (int ti = 0; ti < 2; ++ti)
#pragma unroll
          for (int tj = 0; tj < 2; ++tj)
#pragma unroll
            for (int v = 0; v < 8; ++v)
              S[(ti * 16 + v + 8 * lhalf) * 32 + tj * 16 + l16] = c[ti][tj][v];
        WAVE_LDS_FENCE();

        // P = P * M via V_WMMA_F32_16X16X4_F32 (K truncated at 20 -> 5 k-steps)
#pragma unroll
        for (int ti = 0; ti < 2; ++ti) {
          v2f af[5];
#pragma unroll
          for (int kk = 0; kk < 5; ++kk) {
            const int row = ti * 16 + l16;
            const int kb  = kk * 4 + 2 * lhalf;
            af[kk] = *(const v2f*)(S + row * 32 + kb);
          }
#pragma unroll
          for (int tj = 0; tj < 2; ++tj) {
            v8f acc = {0.f, 0.f, 0.f, 0.f, 0.f, 0.f, 0.f, 0.f};
#pragma unroll
            for (int kk = 0; kk < 5; ++kk) {
              acc = __builtin_amdgcn_wmma_f32_16x16x4_f32(
                  false, af[kk], false, bf[tj][kk], (short)0, acc, false, false);
            }
            c[ti][tj] = acc;
          }
        }
        WAVE_LDS_FENCE();
      }
    }

    // Write block product (rows 0..19; cols >= 20 are exactly zero)
    float* P = prod[side * NBLK + w];
#pragma unroll
    for (int ti = 0; ti < 2; ++ti)
#pragma unroll
      for (int tj = 0; tj < 2; ++tj)
#pragma unroll
        for (int v = 0; v < 8; ++v) {
          const int m = ti * 16 + v + 8 * lhalf;
          if (m < D_) P[m * 32 + tj * 16 + l16] = c[ti][tj][v];
        }
  }
  // Real cross-wave dependency: wave 0 consumes every wave's block product.
  __syncthreads();

  // Phase 2: wave 0 folds the 32 block products with the boundary vectors
  if (tid < 32) {
    // left scan: v <- v * P_j
    float vL = (lane == 0) ? 1.f : 0.f;
    for (int j = 0; j < NBLK; ++j) {
      const float* P = prod[j];
      float nv = 0.f;
      for (int l = 0; l < D_; ++l)
        nv += __shfl(vL, l) * P[l * 32 + lane];
      vL = nv;
    }
    // right scan: v <- P_j * v  (reverse order)
    float vR = (lane == 0) ? 1.f : 0.f;
    for (int j = NBLK - 1; j >= 0; --j) {
      const float* P = prod[NBLK + j];
      const int lrow = (lane < D_) ? lane : 0;
      float nv = 0.f;
      for (int r = 0; r < D_; ++r) {
        float pv = P[lrow * 32 + r];
        pv = (lane < D_) ? pv : 0.f;
        nv += pv * __shfl(vR, r);
      }
      vR = nv;
    }
    // out[b,o] = sum_{l,r} lv[l] * W[o,l,r] * rv[r]
    float lvv[D_], rvv[D_];
#pragma unroll
    for (int i = 0; i < D_; ++i) { lvv[i] = __shfl(vL, i); rvv[i] = __shfl(vR, i); }
    const int o = (lane < O_) ? lane : 0;
    float acc = 0.f;
    for (int l = 0; l < D_; ++l) {
      float t = 0.f;
      for (int r = 0; r < D_; ++r)
        t = fmaf(ot[(o * D_ + l) * D_ + r], rvv[r], t);
      acc = fmaf(lvv[l], t, acc);
    }
    if (lane < O_) out[(size_t)b * O_ + lane] = acc;
  }
}

extern "C" void kernel_launch(void* const* d_in, const int* in_sizes, int n_in,
                              void* d_out, int out_size, void* d_ws, size_t ws_size,
                              hipStream_t stream) {
  const float* inp = (const float*)d_in[0];
  const float* lt  = (const float*)d_in[1];
  const float* ot  = (const float*)d_in[2];
  const float* rt  = (const float*)d_in[3];
  float* out       = (float*)d_out;
  (void)in_sizes; (void)n_in; (void)out_size; (void)d_ws; (void)ws_size;
  mps_fused<<<B_, THREADS, 0, stream>>>(inp, lt, ot, rt, out);
}